// MultiPatchGCN_34514357191316
// MI455X (gfx1250) — compile-verified
//
#include <hip/hip_runtime.h>

typedef __attribute__((ext_vector_type(2))) float v2f;
typedef __attribute__((ext_vector_type(8))) float v8f;

#define NEG_SLOPE 0.01f
#define TM 48            // A-panel rows per block (3 m-tiles)
#define LDSW 260         // padded LDS row stride in floats (bank-conflict-free b64 reads)

__device__ __forceinline__ v8f wmma4(v2f a, v2f b, v8f c) {
  return __builtin_amdgcn_wmma_f32_16x16x4_f32(false, a, false, b, (short)0, c, false, false);
}

// async global -> LDS copy of 16 bytes (per lane), CDNA5 ASYNCcnt path
__device__ __forceinline__ void async_copy16(unsigned lds_off, const float* gsrc) {
  unsigned long long ga = (unsigned long long)gsrc;
  asm volatile("global_load_async_to_lds_b128 %0, %1, off"
               :: "v"(lds_off), "v"(ga) : "memory");
}
__device__ __forceinline__ void async_wait0() {
  asm volatile("s_wait_asynccnt 0x0" ::: "memory");
}

// ---------------------------------------------------------------- utilities
__global__ void zero_kernel(float* __restrict__ p, long long n) {
  long long i = (long long)blockIdx.x * blockDim.x + threadIdx.x;
  long long stride = (long long)gridDim.x * blockDim.x;
  for (; i < n; i += stride) p[i] = 0.0f;
}

__global__ void degree_kernel(const int* __restrict__ src, const int* __restrict__ dst,
                              float* __restrict__ in_deg, float* __restrict__ out_deg, int E) {
  int i = blockIdx.x * blockDim.x + threadIdx.x;
  int stride = gridDim.x * blockDim.x;
  for (; i < E; i += stride) {
    atomicAdd(&in_deg[dst[i]], 1.0f);
    atomicAdd(&out_deg[src[i]], 1.0f);
  }
}

__global__ void scale_kernel(const float* __restrict__ in_deg, const float* __restrict__ out_deg,
                             float* __restrict__ inscale, float* __restrict__ outscale, int N) {
  int i = blockIdx.x * blockDim.x + threadIdx.x;
  int stride = gridDim.x * blockDim.x;
  for (; i < N; i += stride) {
    inscale[i]  = rsqrtf(fmaxf(in_deg[i], 1.0f));
    outscale[i] = rsqrtf(fmaxf(out_deg[i], 1.0f));
  }
}

// q[n] = sum over edges with src==n of in_deg[dst]^{-1/2}
__global__ void qsum_kernel(const int* __restrict__ src, const int* __restrict__ dst,
                            const float* __restrict__ inscale, float* __restrict__ q, int E) {
  int i = blockIdx.x * blockDim.x + threadIdx.x;
  int stride = gridDim.x * blockDim.x;
  for (; i < E; i += stride) atomicAdd(&q[src[i]], inscale[dst[i]]);
}

// agg[dst[e]] += feat[src[e]]   (feature dim = 4 << d4shift floats, float4-vectorized)
__global__ void edge_agg_kernel(const int* __restrict__ src, const int* __restrict__ dst,
                                const float* __restrict__ feat, float* __restrict__ agg,
                                int E, int d4shift) {
  long long total = (long long)E << d4shift;
  long long i = (long long)blockIdx.x * blockDim.x + threadIdx.x;
  long long stride = (long long)gridDim.x * blockDim.x;
  int d4 = 1 << d4shift;
  for (; i < total; i += stride) {
    int e = (int)(i >> d4shift);
    int c = (int)(i & (d4 - 1));
    int s = src[e], d = dst[e];
    float4 v = ((const float4*)feat)[(long long)s * d4 + c];
    float* dp = agg + ((long long)d * d4 + c) * 4;
    atomicAdd(dp + 0, v.x);
    atomicAdd(dp + 1, v.y);
    atomicAdd(dp + 2, v.z);
    atomicAdd(dp + 3, v.w);
  }
}

// ------------------------------------------------------- WMMA GEMM layer 1
// hscaled = leaky_relu(n_feat @ Wself + (agg1 * in_deg^{-1}) @ Wneigh + b) * out_deg^{-1/2}
// Block: 48 A-rows staged in LDS (async), 8 waves x (3 m-tiles x 2 n-tiles) accumulators.
__global__ __launch_bounds__(256) void sage_gemm_kernel(
    const float* __restrict__ nfeat,   // [N,128]
    const float* __restrict__ agg1,    // [N,128]
    const float* __restrict__ Wself,   // [128,256] row-major
    const float* __restrict__ Wneigh,  // [128,256]
    const float* __restrict__ bias,    // [256]
    const float* __restrict__ inscale, // [N]
    const float* __restrict__ outscale,// [N]
    float* __restrict__ hscaled,       // [N,256]
    int N) {
  __shared__ float As[TM * LDSW];      // 48 x 256 (padded), cols 0..127=nfeat, 128..255=agg1
  const int tid = threadIdx.x;
  const int m0 = blockIdx.x * TM;

  // ---- async stage A panel: 48*64 float4 chunks ----
  for (int ci = tid; ci < TM * 64; ci += 256) {
    int row = ci >> 6;
    int c4  = ci & 63;
    int gm = m0 + row; if (gm >= N) gm = N - 1;
    const float* gsrc = (c4 < 32) ? (nfeat + (long long)gm * 128 + c4 * 4)
                                  : (agg1  + (long long)gm * 128 + (c4 - 32) * 4);
    unsigned loff = (unsigned)(unsigned long long)&As[row * LDSW + c4 * 4];
    async_copy16(loff, gsrc);
  }
  async_wait0();
  __syncthreads();

  const int lane = tid & 31;
  const int wave = tid >> 5;
  const int half = lane >> 4;          // 0 | 1
  const int l15  = lane & 15;
  const int koff = half * 2;           // A/B regs hold K = koff + {0,1}
  const int n0   = wave * 32 + l15;    // wave owns cols [wave*32, wave*32+32)

  float invin[3];
  #pragma unroll
  for (int mt = 0; mt < 3; ++mt) {
    int r = m0 + mt * 16 + l15; if (r >= N) r = N - 1;
    float is = inscale[r];
    invin[mt] = is * is;               // == 1/max(in_deg,1)
  }

  v8f c[6];                            // [mt][nt]
  #pragma unroll
  for (int t = 0; t < 6; ++t) c[t] = (v8f){};

  // half 1: K 0..127 (nfeat part), B = Wself
  for (int k = 0; k < 128; k += 4) {
    const float* wr = Wself + (k + koff) * 256;
    v2f bA = { wr[n0],      wr[256 + n0] };
    v2f bB = { wr[n0 + 16], wr[256 + n0 + 16] };
    #pragma unroll
    for (int mt = 0; mt < 3; ++mt) {
      v2f a = *(const v2f*)&As[(mt * 16 + l15) * LDSW + k + koff];
      c[mt * 2 + 0] = wmma4(a, bA, c[mt * 2 + 0]);
      c[mt * 2 + 1] = wmma4(a, bB, c[mt * 2 + 1]);
    }
  }
  // half 2: K 128..255 (mean-neighbor part), B = Wneigh
  for (int k = 128; k < 256; k += 4) {
    const float* wr = Wneigh + (k - 128 + koff) * 256;
    v2f bA = { wr[n0],      wr[256 + n0] };
    v2f bB = { wr[n0 + 16], wr[256 + n0 + 16] };
    #pragma unroll
    for (int mt = 0; mt < 3; ++mt) {
      v2f a = *(const v2f*)&As[(mt * 16 + l15) * LDSW + k + koff];
      a.x *= invin[mt]; a.y *= invin[mt];
      c[mt * 2 + 0] = wmma4(a, bA, c[mt * 2 + 0]);
      c[mt * 2 + 1] = wmma4(a, bB, c[mt * 2 + 1]);
    }
  }

  #pragma unroll
  for (int mt = 0; mt < 3; ++mt) {
    int rbase = m0 + mt * 16 + half * 8;     // C/D layout: lanes>=16 -> M+8
    #pragma unroll
    for (int nt = 0; nt < 2; ++nt) {
      int n = n0 + nt * 16;
      float bn = bias[n];
      v8f cc = c[mt * 2 + nt];
      #pragma unroll
      for (int v = 0; v < 8; ++v) {
        int row = rbase + v;
        if (row < N) {
          float x = cc[v] + bn;
          x = (x >= 0.0f) ? x : NEG_SLOPE * x;
          hscaled[(long long)row * 256 + n] = x * outscale[row];
        }
      }
    }
  }
}

// ------------------------------------------------------- WMMA GEMM layer 2
// h2w = leaky_relu((agg2 * in_deg^{-1/2}) @ W1 + b1) * (out_deg^{-1/2} * q)
__global__ __launch_bounds__(256) void gc_gemm_kernel(
    const float* __restrict__ agg2,    // [N,256]
    const float* __restrict__ W1,      // [256,256]
    const float* __restrict__ bias,    // [256]
    const float* __restrict__ inscale,
    const float* __restrict__ outscale,
    const float* __restrict__ q,
    float* __restrict__ h2w,           // [N,256]
    int N) {
  __shared__ float As[TM * LDSW];
  const int tid = threadIdx.x;
  const int m0 = blockIdx.x * TM;

  for (int ci = tid; ci < TM * 64; ci += 256) {
    int row = ci >> 6;
    int c4  = ci & 63;
    int gm = m0 + row; if (gm >= N) gm = N - 1;
    const float* gsrc = agg2 + (long long)gm * 256 + c4 * 4;
    unsigned loff = (unsigned)(unsigned long long)&As[row * LDSW + c4 * 4];
    async_copy16(loff, gsrc);
  }
  async_wait0();
  __syncthreads();

  const int lane = tid & 31;
  const int wave = tid >> 5;
  const int half = lane >> 4;
  const int l15  = lane & 15;
  const int koff = half * 2;
  const int n0   = wave * 32 + l15;

  float isc[3];
  #pragma unroll
  for (int mt = 0; mt < 3; ++mt) {
    int r = m0 + mt * 16 + l15; if (r >= N) r = N - 1;
    isc[mt] = inscale[r];
  }

  v8f c[6];
  #pragma unroll
  for (int t = 0; t < 6; ++t) c[t] = (v8f){};

  for (int k = 0; k < 256; k += 4) {
    const float* wr = W1 + (k + koff) * 256;
    v2f bA = { wr[n0],      wr[256 + n0] };
    v2f bB = { wr[n0 + 16], wr[256 + n0 + 16] };
    #pragma unroll
    for (int mt = 0; mt < 3; ++mt) {
      v2f a = *(const v2f*)&As[(mt * 16 + l15) * LDSW + k + koff];
      a.x *= isc[mt]; a.y *= isc[mt];
      c[mt * 2 + 0] = wmma4(a, bA, c[mt * 2 + 0]);
      c[mt * 2 + 1] = wmma4(a, bB, c[mt * 2 + 1]);
    }
  }

  #pragma unroll
  for (int mt = 0; mt < 3; ++mt) {
    int rbase = m0 + mt * 16 + half * 8;
    #pragma unroll
    for (int nt = 0; nt < 2; ++nt) {
      int n = n0 + nt * 16;
      float bn = bias[n];
      v8f cc = c[mt * 2 + nt];
      #pragma unroll
      for (int v = 0; v < 8; ++v) {
        int row = rbase + v;
        if (row < N) {
          float x = cc[v] + bn;
          x = (x >= 0.0f) ? x : NEG_SLOPE * x;
          h2w[(long long)row * 256 + n] = x * (outscale[row] * q[row]);
        }
      }
    }
  }
}

// s[k] = sum_n h2w[n][k]
__global__ void colsum_kernel(const float* __restrict__ h2w, float* __restrict__ s, int N, int H) {
  int k = threadIdx.x;
  if (k >= H) return;
  float acc = 0.0f;
  for (int row = blockIdx.x; row < N; row += gridDim.x)
    acc += h2w[(long long)row * H + k];
  atomicAdd(&s[k], acc);
}

// out[0..O-1] = (s/N) @ Wo + bo ; out[O..O+Do-1] = (s/N) @ Wd + bd
__global__ void heads_kernel(const float* __restrict__ s,
                             const float* __restrict__ Wo, const float* __restrict__ bo,
                             const float* __restrict__ Wd, const float* __restrict__ bd,
                             float* __restrict__ out, float invN, int H, int O, int Do) {
  int j = threadIdx.x;
  if (j < O) {
    float acc = 0.0f;
    for (int k = 0; k < H; ++k) acc += s[k] * Wo[k * O + j];
    out[j] = acc * invN + bo[j];
  } else if (j < O + Do) {
    int jj = j - O;
    float acc = 0.0f;
    for (int k = 0; k < H; ++k) acc += s[k] * Wd[k * Do + jj];
    out[O + jj] = acc * invN + bd[jj];
  }
}

// ---------------------------------------------------------------- launcher
extern "C" void kernel_launch(void* const* d_in, const int* in_sizes, int n_in,
                              void* d_out, int out_size, void* d_ws, size_t ws_size,
                              hipStream_t stream) {
  const float* nfeat  = (const float*)d_in[0];
  const int*   src    = (const int*)d_in[1];
  const int*   dst    = (const int*)d_in[2];
  const float* Wself  = (const float*)d_in[3];
  const float* Wneigh = (const float*)d_in[4];
  const float* bsage  = (const float*)d_in[5];
  const float* W1     = (const float*)d_in[6];
  const float* b1     = (const float*)d_in[7];
  const float* Wo     = (const float*)d_in[8];
  const float* bo     = (const float*)d_in[9];
  const float* Wd     = (const float*)d_in[10];
  const float* bd     = (const float*)d_in[11];
  float* out = (float*)d_out;

  const int H   = in_sizes[5];            // 256
  const int Din = in_sizes[3] / H;        // 128
  const int N   = in_sizes[0] / Din;      // 50000
  const int E   = in_sizes[1];            // 800000
  const int O   = in_sizes[9];            // 100
  const int Do  = in_sizes[11];           // 8

  float* ws       = (float*)d_ws;
  float* in_deg   = ws;                            // N
  float* out_deg  = in_deg + N;                    // N
  float* inscale  = out_deg + N;                   // N
  float* outscale = inscale + N;                   // N
  float* q        = outscale + N;                  // N
  float* s        = q + N;                         // 256
  float* agg1     = s + 256;                       // N*128
  float* hscaled  = agg1 + (long long)N * Din;     // N*256 (reused as h2w)
  float* agg2     = hscaled + (long long)N * H;    // N*256

  const int T = 256;
  long long zero1 = 5LL * N + 256 + (long long)N * Din;  // header + agg1
  long long zero2 = (long long)N * H;                    // agg2

  zero_kernel<<<2048, T, 0, stream>>>(ws, zero1);
  zero_kernel<<<2048, T, 0, stream>>>(agg2, zero2);

  int eBlocks = (E + T - 1) / T;
  degree_kernel<<<eBlocks, T, 0, stream>>>(src, dst, in_deg, out_deg, E);
  scale_kernel<<<(N + T - 1) / T, T, 0, stream>>>(in_deg, out_deg, inscale, outscale, N);
  qsum_kernel<<<eBlocks, T, 0, stream>>>(src, dst, inscale, q, E);

  {
    long long total = (long long)E * (Din / 4);   // shift 5
    int blocks = (int)((total + T - 1) / T);
    edge_agg_kernel<<<blocks, T, 0, stream>>>(src, dst, nfeat, agg1, E, 5);
  }

  int gemmBlocks = (N + TM - 1) / TM;
  sage_gemm_kernel<<<gemmBlocks, T, 0, stream>>>(nfeat, agg1, Wself, Wneigh, bsage,
                                                 inscale, outscale, hscaled, N);

  {
    long long total = (long long)E * (H / 4);     // shift 6
    int blocks = (int)((total + T - 1) / T);
    edge_agg_kernel<<<blocks, T, 0, stream>>>(src, dst, hscaled, agg2, E, 6);
  }

  gc_gemm_kernel<<<gemmBlocks, T, 0, stream>>>(agg2, W1, b1, inscale, outscale, q,
                                               hscaled /* h2w */, N);

  colsum_kernel<<<256, T, 0, stream>>>(hscaled, s, N, H);
  heads_kernel<<<1, 128, 0, stream>>>(s, Wo, bo, Wd, bd, out, 1.0f / (float)N, H, O, Do);
}